// NCELoss_46162308497609
// MI455X (gfx1250) — compile-verified
//
#include <hip/hip_runtime.h>
#include <hip/hip_bf16.h>
#include <cstdint>

// Problem constants from the reference.
#define BATCH   4
#define SEQ     512
#define NROWS   (BATCH * SEQ)     // 2048
#define VOCAB   50257
#define V_PAD   50272             // multiple of 16 (fine scan reads float4 x4)
#define G_TOT   3142              // ceil(VOCAB/16)
#define G_PAD   3152              // multiple of 4, >= G_TOT
#define NCHUNK  50                // ceil(V_PAD/1024); 50*1024 = 51200

// ws layout (floats): [0..V_PAD) cdf | [V_PAD..V_PAD+G_PAD) ccdf |
//                     [+0..64) chunk sums | [+64..128) chunk offsets (+ total at [NCHUNK])

// ---------- helpers ----------

__device__ __forceinline__ uint32_t wang_hash(uint32_t x) {
    x = (x ^ 61u) ^ (x >> 16);
    x *= 9u;
    x ^= x >> 4;
    x *= 0x27d4eb2du;
    x ^= x >> 15;
    return x;
}

__device__ __forceinline__ float log_sigmoidf(float x) {
    // log(sigmoid(x)) = min(x,0) - log1p(exp(-|x|))
    return fminf(x, 0.0f) - log1pf(__expf(-fabsf(x)));
}

// CDNA5 async global -> LDS copy (16 bytes per lane), tracked by ASYNCcnt.
__device__ __forceinline__ void async_copy_b128(uint32_t lds_byte_addr, const void* gaddr) {
    asm volatile("global_load_async_to_lds_b128 %0, %1, off"
                 :: "v"(lds_byte_addr), "v"(gaddr)
                 : "memory");
}

__device__ __forceinline__ void wait_asynccnt0() {
    asm volatile("s_wait_asynccnt 0" ::: "memory");
}

#if defined(__has_builtin)
#if __has_builtin(__builtin_amdgcn_tensor_load_to_lds) && __has_builtin(__builtin_amdgcn_s_wait_tensorcnt)
#define USE_TDM 1
#endif
#endif

#ifdef USE_TDM
typedef unsigned int u32x4 __attribute__((ext_vector_type(4)));
typedef int          i32x4 __attribute__((ext_vector_type(4)));
typedef int          i32x8 __attribute__((ext_vector_type(8)));

// One-descriptor TDM DMA: contiguous 1-D tile of `nelem` f32 from global -> LDS.
// 6-arg builtin form (clang-23 / therock-10.0 toolchain).
__device__ __forceinline__ void tdm_load_1d_f32(uint32_t lds_byte_addr, const void* gaddr, int nelem) {
    const uint64_t ga = (uint64_t)(uintptr_t)gaddr;
    u32x4 g0;
    g0.x = 1u;                                        // count=1, user descriptor
    g0.y = lds_byte_addr;                             // lds_addr
    g0.z = (uint32_t)ga;                              // global_addr[31:0]
    g0.w = (uint32_t)(ga >> 32) | (2u << 30);         // global_addr[56:32] | type=2 ("image")
    i32x8 g1;
    g1[0] = (2 << 16);                                // workgroup_mask=0, data_size=4B
    g1[1] = (nelem & 0xFFFF) << 16;                   // tensor_dim0 [low16 -> bits 63:48]
    g1[2] = ((unsigned)nelem >> 16) | (1u << 16);     // tensor_dim0 hi16 | tensor_dim1=1 lo16
    g1[3] = (nelem & 0xFFFF) << 16;                   // tensor_dim1 hi16=0 | tile_dim0=nelem
    g1[4] = 1;                                        // tile_dim1=1, tile_dim2=0
    g1[5] = nelem;                                    // tensor_dim0_stride[31:0]
    g1[6] = (nelem & 0xFFFF) << 16;                   // dim0_stride hi=0 | dim1_stride lo16
    g1[7] = 0;                                        // dim1_stride hi
    i32x4 gz4 = {0, 0, 0, 0};
    i32x8 gz8 = {0, 0, 0, 0, 0, 0, 0, 0};
    __builtin_amdgcn_tensor_load_to_lds(g0, g1, gz4, gz4, gz8, 0);
}
#endif

// ---------- kernel A: per-chunk local inclusive scan ----------
__global__ __launch_bounds__(1024)
void scan_local_kernel(const float* __restrict__ noise,
                       float* __restrict__ cdf,          // [V_PAD] chunk-local prefix
                       float* __restrict__ chunk_sums) { // [NCHUNK]
    __shared__ float wsum[32];
    const int tid  = threadIdx.x;
    const int lane = tid & 31, wave = tid >> 5;
    const int i    = blockIdx.x * 1024 + tid;

    float v = (i < VOCAB) ? noise[i] : 0.0f;
    // wave32 inclusive scan (register-only)
    #pragma unroll
    for (int off = 1; off < 32; off <<= 1) {
        float t = __shfl_up(v, off, 32);
        if (lane >= off) v += t;
    }
    if (lane == 31) wsum[wave] = v;
    __syncthreads();
    if (wave == 0) {                     // scan the 32 wave sums
        float s = wsum[lane];
        #pragma unroll
        for (int off = 1; off < 32; off <<= 1) {
            float t = __shfl_up(s, off, 32);
            if (lane >= off) s += t;
        }
        wsum[lane] = s;
    }
    __syncthreads();
    const float incl = v + ((wave > 0) ? wsum[wave - 1] : 0.0f);
    if (i < V_PAD) cdf[i] = incl;        // pads carry the chunk total (v adds zeros)
    if (tid == 0) chunk_sums[blockIdx.x] = wsum[31];
}

// ---------- kernel B: scan the 50 chunk sums; zero output ----------
__global__ void scan_chunks_kernel(const float* __restrict__ chunk_sums,
                                   float* __restrict__ chunk_off,  // [NCHUNK+1], [NCHUNK]=total
                                   float* __restrict__ out) {
    if (threadIdx.x == 0) {
        float run = 0.0f;
        for (int c = 0; c < NCHUNK; ++c) { chunk_off[c] = run; run += chunk_sums[c]; }
        chunk_off[NCHUNK] = run;          // total
        out[0] = 0.0f;
    }
}

// ---------- kernel C: fixup to global CDF + build coarse table ----------
__global__ __launch_bounds__(1024)
void scan_fixup_kernel(float* __restrict__ cdf,
                       float* __restrict__ ccdf,
                       const float* __restrict__ chunk_off) {
    const int i     = blockIdx.x * 1024 + threadIdx.x;   // up to 51199
    const float off = chunk_off[blockIdx.x];
    const float tot = chunk_off[NCHUNK];
    const float val = (i < VOCAB) ? (cdf[i] + off) : tot;
    if (i < V_PAD) cdf[i] = val;                          // pads == total
    if ((i & 15) == 15) {
        const int g = i >> 4;
        if (g < G_PAD) ccdf[g] = val;                     // pads == total
    }
}

// ---------- kernel D: sampling + scoring ----------
__global__ __launch_bounds__(256)
void nce_loss_kernel(const float* __restrict__ logits,      // [NROWS, VOCAB]
                     const long long* __restrict__ target,  // [NROWS] int64
                     const int* __restrict__ kptr,          // scalar k
                     const float* __restrict__ cdf,         // [V_PAD]
                     const float* __restrict__ ccdf_g,      // [G_PAD]
                     float* __restrict__ out) {
    __shared__ __align__(16) float ccdf_s[G_PAD];   // 12608 B
    __shared__ float red[8];

    const int tid = threadIdx.x;
    const int row = blockIdx.x;
    const int k   = kptr[0];

    // Stage coarse CDF into LDS: TDM DMA if available, else ASYNCcnt copies.
#ifdef USE_TDM
    if (tid < 32) {     // one wave issues the DMA descriptor
        tdm_load_1d_f32((uint32_t)(uintptr_t)(&ccdf_s[0]), ccdf_g, G_PAD);
        __builtin_amdgcn_s_wait_tensorcnt(0);
    }
    __syncthreads();
#else
    {
        const uint32_t lds_base = (uint32_t)(uintptr_t)(&ccdf_s[0]);
        for (int i = tid; i < G_PAD / 4; i += 256)
            async_copy_b128(lds_base + 16u * (uint32_t)i, ccdf_g + 4 * i);
        wait_asynccnt0();
        __syncthreads();
    }
#endif

    const float total = cdf[VOCAB - 1];             // L2-resident, uniform
    const float u     = (wang_hash((uint32_t)row) + 0.5f) * 2.3283064e-10f;
    const float scale = total / (float)k;
    const float* rowp = logits + (size_t)row * (size_t)VOCAB;

    float acc = 0.0f;

    // Stratified inverse-CDF sampling: coarse search in LDS + fine float4 probe in L2.
    for (int j = tid; j < k; j += 256) {
        const float t = ((float)j + u) * scale;     // in [0, total)

        int lo = 0, hi = G_TOT - 1;                 // first g with ccdf[g] >= t
        while (lo < hi) {
            const int mid = (lo + hi) >> 1;
            if (ccdf_s[mid] < t) lo = mid + 1; else hi = mid;
        }
        const int g    = lo;
        const int idx0 = g << 4;                    // 64B-aligned group start

        const float4* grp = (const float4*)(cdf + idx0);
        const float4 a0 = grp[0], a1 = grp[1], a2 = grp[2], a3 = grp[3];
        float vals[16] = {a0.x,a0.y,a0.z,a0.w, a1.x,a1.y,a1.z,a1.w,
                          a2.x,a2.y,a2.z,a2.w, a3.x,a3.y,a3.z,a3.w};

        float c_prev = (g > 0) ? ccdf_s[g - 1] : 0.0f;
        int   idx    = -1;
        float p_raw  = 0.0f;
        #pragma unroll
        for (int r = 0; r < 16; ++r) {
            const float c = vals[r];
            const bool last = (idx0 + r >= VOCAB - 1);
            if (idx < 0 && (c >= t || last)) {
                idx = (idx0 + r < VOCAB) ? (idx0 + r) : (VOCAB - 1);
                p_raw = c - c_prev;
            }
            if (idx < 0) c_prev = c;
        }
        p_raw = fmaxf(p_raw, 1e-30f);

        const float score = rowp[idx];
        const float delta = score - __logf((float)k * p_raw / total);
        acc += log_sigmoidf(-delta);
    }

    // Target term (one lane).
    if (tid == 0) {
        long long tl = target[row];
        int tgt = (int)tl;
        tgt = (tgt < 0) ? 0 : ((tgt >= VOCAB) ? VOCAB - 1 : tgt);
        const float c1  = cdf[tgt];
        const float c0  = (tgt > 0) ? cdf[tgt - 1] : 0.0f;
        const float p_t = fmaxf(c1 - c0, 1e-30f);
        const float delta_t = rowp[tgt] - __logf((float)k * p_t / total);
        acc += log_sigmoidf(delta_t);
    }

    // Block reduction: wave32 shuffles, then cross-wave via LDS.
    for (int off = 16; off > 0; off >>= 1) acc += __shfl_down(acc, off, 32);
    const int wave = tid >> 5, lane = tid & 31;
    if (lane == 0) red[wave] = acc;
    __syncthreads();
    if (tid == 0) {
        float s = 0.0f;
        #pragma unroll
        for (int w = 0; w < 8; ++w) s += red[w];
        atomicAdd(out, -s * (1.0f / (float)NROWS));   // nce = -(...), mean over rows
    }
}

// ---------- launcher ----------
extern "C" void kernel_launch(void* const* d_in, const int* in_sizes, int n_in,
                              void* d_out, int out_size, void* d_ws, size_t ws_size,
                              hipStream_t stream) {
    const float*     logits = (const float*)d_in[0];       // [B,T,V] f32
    const float*     noise  = (const float*)d_in[1];       // [V] f32
    const long long* target = (const long long*)d_in[2];   // [B,T] i64
    const int*       kptr   = (const int*)d_in[3];         // scalar
    float*           out    = (float*)d_out;               // [1] f32

    float* cdf   = (float*)d_ws;                           // V_PAD floats
    float* ccdf  = cdf + V_PAD;                            // G_PAD floats
    float* sums  = ccdf + G_PAD;                           // NCHUNK floats
    float* offs  = sums + 64;                              // NCHUNK+1 floats

    scan_local_kernel <<<NCHUNK, 1024, 0, stream>>>(noise, cdf, sums);
    scan_chunks_kernel<<<1, 32, 0, stream>>>(sums, offs, out);
    scan_fixup_kernel <<<NCHUNK, 1024, 0, stream>>>(cdf, ccdf, offs);
    nce_loss_kernel   <<<NROWS, 256, 0, stream>>>(logits, target, kptr, cdf, ccdf, out);

    (void)in_sizes; (void)n_in; (void)out_size; (void)ws_size;
}